// SpGraphAttentionLayer_60696477827762
// MI455X (gfx1250) — compile-verified
//
#include <hip/hip_runtime.h>
#include <hip/hip_bf16.h>

#define D 128
#define GAT_ALPHA 0.2f
#define GAT_LN_EPS 1e-5f

typedef float v2f __attribute__((ext_vector_type(2)));
typedef float v8f __attribute__((ext_vector_type(8)));

// -----------------------------------------------------------------------------
// Kernel 1: h = x @ W^T + b  (fp32 WMMA 16x16x4), fused s1 = h@a1, s2 = h@a2.
// One block = 256 threads = 8 waves; block computes a 16-row x 128-col tile of h.
// Wave w computes columns [16w, 16w+16) with 32 chained V_WMMA_F32_16X16X4_F32.
// x tile staged through LDS (stride 132 to avoid bank conflicts).
// -----------------------------------------------------------------------------
__global__ __launch_bounds__(256)
void gat_gemm_kernel(const float* __restrict__ x, const float* __restrict__ W,
                     const float* __restrict__ bias, const float* __restrict__ avec,
                     float* __restrict__ h, float* __restrict__ s1,
                     float* __restrict__ s2, int N)
{
    __shared__ float xs[16 * 132];
    __shared__ float hs[16 * 132];

    const int tid     = threadIdx.x;
    const int wave    = tid >> 5;
    const int lane    = tid & 31;
    const int m       = lane & 15;   // row (A) / col (B) within 16
    const int hi      = lane >> 4;   // K-pair select per ISA layout
    const int rowbase = blockIdx.x * 16;
    const int colbase = wave * 16;

    // Cooperative, coalesced load of the 16x128 x-tile into LDS (clamped tail).
    for (int idx = tid; idx < 16 * D; idx += 256) {
        int r = idx >> 7, c = idx & 127;
        int row = rowbase + r; if (row >= N) row = N - 1;
        xs[r * 132 + c] = x[(size_t)row * D + c];
    }
    __syncthreads();

    v8f acc = {};
    #pragma unroll
    for (int k0 = 0; k0 < D; k0 += 4) {
        const int kb = k0 + 2 * hi;               // lanes 16-31 hold K+2 (ISA layout)
        v2f af, bf;
        af[0] = xs[m * 132 + kb];                 // A: 16x4 tile of x
        af[1] = xs[m * 132 + kb + 1];
        const float2 wv = *(const float2*)(W + (size_t)(colbase + m) * D + kb);
        bf[0] = wv.x;                             // B: 4x16 tile of W^T (= W rows)
        bf[1] = wv.y;
        acc = __builtin_amdgcn_wmma_f32_16x16x4_f32(
            /*neg_a=*/false, af, /*neg_b=*/false, bf,
            /*c_mod=*/(short)0, acc, /*reuse_a=*/false, /*reuse_b=*/false);
    }

    // Epilogue: bias add, write h, stage tile to LDS for the score reduction.
    #pragma unroll
    for (int r = 0; r < 8; ++r) {
        const int rl  = r + 8 * hi;               // C layout: VGPR r -> M=r / M=8+r
        const int row = rowbase + rl;
        const float val = acc[r] + bias[colbase + m];
        hs[rl * 132 + colbase + m] = val;
        if (row < N) h[(size_t)row * D + colbase + m] = val;
    }
    __syncthreads();

    // s1/s2: wave w reduces rows {2w, 2w+1}; each 16-lane half owns one row,
    // each lane covers 8 columns, then a 16-lane shfl_xor tree reduction.
    const int srow = 2 * wave + hi;
    const float* hr = &hs[srow * 132];
    float p1 = 0.f, p2 = 0.f;
    #pragma unroll
    for (int j = 0; j < 8; ++j) {
        const float v = hr[m * 8 + j];
        p1 = fmaf(v, avec[m * 8 + j], p1);
        p2 = fmaf(v, avec[D + m * 8 + j], p2);
    }
    #pragma unroll
    for (int off = 8; off >= 1; off >>= 1) {
        p1 += __shfl_xor(p1, off, 32);
        p2 += __shfl_xor(p2, off, 32);
    }
    const int row = rowbase + srow;
    if (m == 0 && row < N) { s1[row] = p1; s2[row] = p2; }
}

// -----------------------------------------------------------------------------
// Kernel 2: zero h_prime (float4 stores).
// -----------------------------------------------------------------------------
__global__ __launch_bounds__(256)
void gat_zero_kernel(float4* __restrict__ p, long n4)
{
    const long i = (long)blockIdx.x * blockDim.x + threadIdx.x;
    if (i < n4) p[i] = make_float4(0.f, 0.f, 0.f, 0.f);
}

// -----------------------------------------------------------------------------
// Kernel 3: edge scatter. One wave32 per edge: coalesced 512B row gather of
// h[dst] (L2-resident), per-lane float4 scale, 4 global_atomic_add_f32 into
// h_prime[src]. Prefetch next edge's row to hide L2 latency.
// -----------------------------------------------------------------------------
__global__ __launch_bounds__(256)
void gat_edge_kernel(const int* __restrict__ src, const int* __restrict__ dst,
                     const float* __restrict__ s1, const float* __restrict__ s2,
                     const float* __restrict__ h, float* __restrict__ hp, int E)
{
    const int lane   = threadIdx.x & 31;
    const int wid    = blockIdx.x * (blockDim.x >> 5) + (threadIdx.x >> 5);
    const int stride = gridDim.x * (blockDim.x >> 5);

    for (int e = wid; e < E; e += stride) {
        const int s = src[e];
        const int d = dst[e];

        // Software prefetch of the next edge's gather row (global_prefetch_b8).
        const int en = e + stride;
        if (en < E) {
            const int dn = dst[en];
            __builtin_prefetch(h + (size_t)dn * D + lane * 4, 0, 0);
        }

        const float z  = s1[s] + s2[d];
        const float lr = (z > 0.f) ? z : GAT_ALPHA * z;
        const float w  = __expf(-lr);

        const float4 hv = *(const float4*)(h + (size_t)d * D + lane * 4);
        float* o = hp + (size_t)s * D + lane * 4;
        atomicAdd(o + 0, w * hv.x);
        atomicAdd(o + 1, w * hv.y);
        atomicAdd(o + 2, w * hv.z);
        atomicAdd(o + 3, w * hv.w);
    }
}

// -----------------------------------------------------------------------------
// Kernel 4: LayerNorm + ELU. One wave32 per row, float4 per lane, shfl_xor
// tree for mean / mean-of-squares.
// -----------------------------------------------------------------------------
__global__ __launch_bounds__(256)
void gat_ln_kernel(const float* __restrict__ hp, const float* __restrict__ lw,
                   const float* __restrict__ lb, float* __restrict__ out, int N)
{
    const int lane = threadIdx.x & 31;
    const int row  = blockIdx.x * (blockDim.x >> 5) + (threadIdx.x >> 5);
    if (row >= N) return;

    const float4 v = *(const float4*)(hp + (size_t)row * D + lane * 4);
    float s = v.x + v.y + v.z + v.w;
    float q = v.x * v.x + v.y * v.y + v.z * v.z + v.w * v.w;
    #pragma unroll
    for (int off = 16; off >= 1; off >>= 1) {
        s += __shfl_xor(s, off, 32);
        q += __shfl_xor(q, off, 32);
    }
    const float mu  = s * (1.f / (float)D);
    const float var = q * (1.f / (float)D) - mu * mu;
    const float rs  = rsqrtf(var + GAT_LN_EPS);

    const float4 w4 = *(const float4*)(lw + lane * 4);
    const float4 b4 = *(const float4*)(lb + lane * 4);
    float4 o;
    {
        float hn;
        hn = (v.x - mu) * rs * w4.x + b4.x; o.x = (hn > 0.f) ? hn : (__expf(hn) - 1.f);
        hn = (v.y - mu) * rs * w4.y + b4.y; o.y = (hn > 0.f) ? hn : (__expf(hn) - 1.f);
        hn = (v.z - mu) * rs * w4.z + b4.z; o.z = (hn > 0.f) ? hn : (__expf(hn) - 1.f);
        hn = (v.w - mu) * rs * w4.w + b4.w; o.w = (hn > 0.f) ? hn : (__expf(hn) - 1.f);
    }
    *(float4*)(out + (size_t)row * D + lane * 4) = o;
}

// -----------------------------------------------------------------------------
extern "C" void kernel_launch(void* const* d_in, const int* in_sizes, int n_in,
                              void* d_out, int out_size, void* d_ws, size_t ws_size,
                              hipStream_t stream) {
    const float* x    = (const float*)d_in[0];
    const float* W    = (const float*)d_in[1];
    const float* bias = (const float*)d_in[2];
    const float* avec = (const float*)d_in[3];
    const float* ln_w = (const float*)d_in[4];
    const float* ln_b = (const float*)d_in[5];
    const int*   edge = (const int*)d_in[6];

    const int N = in_sizes[0] / D;
    const int E = in_sizes[6] / 2;
    const int* src = edge;
    const int* dst = edge + E;

    float* h  = (float*)d_ws;                 // N*128 f32
    float* hp = h  + (size_t)N * D;           // N*128 f32
    float* s1 = hp + (size_t)N * D;           // N f32
    float* s2 = s1 + N;                       // N f32
    float* out = (float*)d_out;

    // 1) GEMM + fused attention scores
    {
        dim3 grid((N + 15) / 16);
        gat_gemm_kernel<<<grid, 256, 0, stream>>>(x, W, bias, avec, h, s1, s2, N);
    }
    // 2) zero h_prime
    {
        const long n4 = (long)N * (D / 4);
        dim3 grid((unsigned)((n4 + 255) / 256));
        gat_zero_kernel<<<grid, 256, 0, stream>>>((float4*)hp, n4);
    }
    // 3) edge scatter (wave per edge, grid-stride)
    {
        dim3 grid(4096);
        gat_edge_kernel<<<grid, 256, 0, stream>>>(src, dst, s1, s2, h, hp, E);
    }
    // 4) LayerNorm + ELU (wave per row)
    {
        dim3 grid((N + 7) / 8);
        gat_ln_kernel<<<grid, 256, 0, stream>>>(hp, ln_w, ln_b, out, N);
    }
}